// deepNet_14224931685023
// MI455X (gfx1250) — compile-verified
//
#include <hip/hip_runtime.h>
#include <hip/hip_bf16.h>

typedef __attribute__((ext_vector_type(2))) float v2f;
typedef __attribute__((ext_vector_type(8))) float v8f;

#define NNODES 50000
#define NEDGES 800000
#define NEG_SLOPE 0.01f

// ---------------------------------------------------------------- utilities
__global__ void fill_zero_kernel(float* __restrict__ p, int n) {
  int i = blockIdx.x * blockDim.x + threadIdx.x;
  if (i < n) p[i] = 0.0f;
}

__global__ void degree_kernel(const int* __restrict__ src, const int* __restrict__ dst,
                              float* __restrict__ degOut, float* __restrict__ degIn, int E) {
  int e = blockIdx.x * blockDim.x + threadIdx.x;
  if (e >= E) return;
  atomicAdd(&degOut[src[e]], 1.0f);
  atomicAdd(&degIn[dst[e]], 1.0f);
}

__global__ void norm_kernel(const float* __restrict__ degOut, const float* __restrict__ degIn,
                            float* __restrict__ nsrc, float* __restrict__ ndst, int n) {
  int i = blockIdx.x * blockDim.x + threadIdx.x;
  if (i >= n) return;
  float dO = degOut[i];
  float dI = degIn[i];
  nsrc[i] = (dO > 0.0f) ? rsqrtf(fmaxf(dO, 1.0f)) : 0.0f;
  ndst[i] = (dI > 0.0f) ? rsqrtf(fmaxf(dI, 1.0f)) : 0.0f;
}

// ------------------------------------------------- edge scatter (SpMM part)
// One thread per (edge, 4-feature chunk). msg = n_src[s] * H[s,:] scattered
// into agg[d,:] with f32 atomics. Node tables are 12.8MB << 192MB L2, so the
// random gathers and the atomic adds resolve in L2, not HBM.
template <bool VEC4>
__global__ void scatter_kernel(const float* __restrict__ H,
                               const int* __restrict__ src, const int* __restrict__ dst,
                               const float* __restrict__ nsrc,
                               float* __restrict__ agg, int E, int F, int CH) {
  int t = blockIdx.x * blockDim.x + threadIdx.x;
  if (t >= E * CH) return;
  int e = t / CH;
  int c = t - e * CH;
  // prefetch upcoming edge indices (global_prefetch_b8 on gfx1250)
  if (e + 2048 < E) {
    __builtin_prefetch(&src[e + 2048], 0, 1);
    __builtin_prefetch(&dst[e + 2048], 0, 1);
  }
  int s = src[e];
  int d = dst[e];
  float ns = nsrc[s];
  int f0 = c * 4;
  const float* hrow = H + (size_t)s * F;
  float* arow = agg + (size_t)d * F;
  if (VEC4) {  // F % 4 == 0: one global_load_b128 per chunk
    float4 h4 = *(const float4*)(hrow + f0);
    atomicAdd(&arow[f0 + 0], h4.x * ns);
    atomicAdd(&arow[f0 + 1], h4.y * ns);
    atomicAdd(&arow[f0 + 2], h4.z * ns);
    atomicAdd(&arow[f0 + 3], h4.w * ns);
  } else {
#pragma unroll
    for (int j = 0; j < 4; ++j) {
      int f = f0 + j;
      if (f < F) atomicAdd(&arow[f], hrow[f] * ns);
    }
  }
}

// ------------------------------------------------------- WMMA f32 GEMM tile
// O[m, 0..FOUT) = act( rowscale[m] * A[m, 0..FIN) @ W + bias ).
// One wave owns a 16 x (NT*16) row-block: the A fragment is loaded once per
// k-step and reused across NT column tiles (NT accumulators), B comes from
// LDS. W is staged LDS-side via gfx1250 async global->LDS copies when the
// width needs no zero padding.
template <int FIN, int FOUT, bool SCALE, bool BIAS, bool RELU>
__global__ void gcn_gemm_wmma(const float* __restrict__ A, const float* __restrict__ W,
                              const float* __restrict__ rowscale, const float* __restrict__ bias,
                              float* __restrict__ O, int mTiles) {
  constexpr int NT = (FOUT + 15) / 16;
  constexpr int FOUTP = NT * 16;
  __shared__ float ldsW[FIN * FOUTP];  // <= 16KB of the 320KB/WGP LDS
  int tid = threadIdx.x;

  if (FOUT == FOUTP) {
    // Async DMA W into LDS: 128 bits per lane, tracked by ASYNCcnt.
    constexpr int NVEC = (FIN * FOUTP) / 4;
    for (int i = tid; i < NVEC; i += blockDim.x) {
      unsigned ldsOff = (unsigned)(uintptr_t)(&ldsW[i * 4]);
      unsigned long long gaddr = (unsigned long long)(uintptr_t)(W + i * 4);
      asm volatile("global_load_async_to_lds_b128 %0, %1, off"
                   :: "v"(ldsOff), "v"(gaddr) : "memory");
    }
    asm volatile("s_wait_asynccnt 0x0" ::: "memory");
  } else {
    // Padded width: scalar fill with zeroed tail columns.
    for (int i = tid; i < FIN * FOUTP; i += blockDim.x) {
      int k = i / FOUTP;
      int c = i - k * FOUTP;
      ldsW[i] = (c < FOUT) ? W[k * FOUT + c] : 0.0f;
    }
  }
  __syncthreads();

  int wave = tid >> 5;
  int lane = tid & 31;
  int tM = blockIdx.x * (blockDim.x >> 5) + wave;
  if (tM >= mTiles) return;  // wave-uniform: EXEC stays all-ones for WMMA

  int half = lane >> 4;  // lane halves carry the K / K+2 planes
  int l15 = lane & 15;
  int m = tM * 16 + l15;  // mTiles*16 == N exactly, no row guard needed
  float scale = SCALE ? rowscale[m] : 1.0f;
  const float* arow = A + (size_t)m * FIN;

  v8f acc[NT] = {};
#pragma unroll
  for (int k = 0; k < FIN; k += 4) {
    v2f a;
    // 32-bit A 16x4 layout: lanes0-15 VGPR{0,1}=K{k,k+1}; lanes16-31 =K{k+2,k+3}
    a.x = arow[k + 2 * half] * scale;
    a.y = arow[k + 2 * half + 1] * scale;
#pragma unroll
    for (int t = 0; t < NT; ++t) {
      v2f b;
      b.x = ldsW[(k + 2 * half) * FOUTP + t * 16 + l15];
      b.y = ldsW[(k + 2 * half + 1) * FOUTP + t * 16 + l15];
      acc[t] = __builtin_amdgcn_wmma_f32_16x16x4_f32(
          /*neg_a=*/false, a, /*neg_b=*/false, b,
          /*c_mod=*/(short)0, acc[t], /*reuse_a=*/false, /*reuse_b=*/false);
    }
  }

#pragma unroll
  for (int t = 0; t < NT; ++t) {
    int col = t * 16 + l15;
    float bv = 0.0f;
    if (BIAS && col < FOUT) bv = bias[col];
#pragma unroll
    for (int r = 0; r < 8; ++r) {
      // C/D layout: VGPR r holds M=r (lanes 0-15) and M=r+8 (lanes 16-31)
      int row = tM * 16 + r + 8 * half;
      float v = acc[t][r] + bv;
      if (RELU) v = (v > 0.0f) ? v : NEG_SLOPE * v;
      if (col < FOUT) O[(size_t)row * FOUT + col] = v;
    }
  }
}

// -------------------------------------------------- layer-5 final epilogue
__global__ void final_bias_kernel(float* __restrict__ out, const float* __restrict__ ndst,
                                  const float* __restrict__ b, int n, int F) {
  int i = blockIdx.x * blockDim.x + threadIdx.x;
  if (i >= n * F) return;
  int node = i / F;
  int f = i - node * F;
  out[i] = out[i] * ndst[node] + b[f];
}

// ---------------------------------------------------------------- launcher
static inline void zero_async(float* p, int n, hipStream_t s) {
  fill_zero_kernel<<<(n + 255) / 256, 256, 0, s>>>(p, n);
}

extern "C" void kernel_launch(void* const* d_in, const int* in_sizes, int n_in,
                              void* d_out, int out_size, void* d_ws, size_t ws_size,
                              hipStream_t stream) {
  (void)in_sizes; (void)n_in; (void)out_size; (void)ws_size;
  const float* feat = (const float*)d_in[0];
  const int* src = (const int*)d_in[1];
  const int* dst = (const int*)d_in[2];
  const float* W1 = (const float*)d_in[3];  const float* b1 = (const float*)d_in[4];
  const float* W2 = (const float*)d_in[5];  const float* b2 = (const float*)d_in[6];
  const float* W3 = (const float*)d_in[7];  const float* b3 = (const float*)d_in[8];
  const float* W4 = (const float*)d_in[9];  const float* b4 = (const float*)d_in[10];
  const float* W5 = (const float*)d_in[11]; const float* b5 = (const float*)d_in[12];

  const int N = NNODES, E = NEDGES;
  float* ws   = (float*)d_ws;
  float* degO = ws;
  float* degI = ws + (size_t)N;
  float* nsrc = ws + (size_t)2 * N;
  float* ndst = ws + (size_t)3 * N;
  float* bufA = ws + (size_t)4 * N;     // N x 64 (aggregation)
  float* bufB = bufA + (size_t)N * 64;  // N x 64 (activations)
  float* out  = (float*)d_out;          // N x 3

  const int mTiles = N / 16;                  // 3125 exactly
  const int gemmBlocks = (mTiles + 3) / 4;    // 4 waves / block, 1 row-block / wave

  // degrees -> symmetric norms
  zero_async(degO, N, stream);
  zero_async(degI, N, stream);
  degree_kernel<<<(E + 255) / 256, 256, 0, stream>>>(src, dst, degO, degI, E);
  norm_kernel<<<(N + 255) / 256, 256, 0, stream>>>(degO, degI, nsrc, ndst, N);

  // Layer 1: propagate raw 4-wide features, then GEMM 4->64 (+ndst, bias, lrelu fused)
  zero_async(bufA, N * 4, stream);
  scatter_kernel<true><<<(E + 255) / 256, 256, 0, stream>>>(feat, src, dst, nsrc, bufA, E, 4, 1);
  gcn_gemm_wmma<4, 64, true, true, true>
      <<<gemmBlocks, 128, 0, stream>>>(bufA, W1, ndst, b1, bufB, mTiles);

  // Layers 2-4: aggregate 64-wide, then GEMM 64->64 (+ndst, bias, lrelu fused)
  const float* Ws[3] = {W2, W3, W4};
  const float* bs[3] = {b2, b3, b4};
  for (int l = 0; l < 3; ++l) {
    zero_async(bufA, N * 64, stream);
    scatter_kernel<true><<<(E * 16 + 255) / 256, 256, 0, stream>>>(bufB, src, dst, nsrc, bufA, E, 64, 16);
    gcn_gemm_wmma<64, 64, true, true, true>
        <<<gemmBlocks, 128, 0, stream>>>(bufA, Ws[l], ndst, bs[l], bufB, mTiles);
  }

  // Layer 5: GEMM 64->3 first, then propagate 3-wide, then *ndst + b5
  gcn_gemm_wmma<64, 3, false, false, false>
      <<<gemmBlocks, 128, 0, stream>>>(bufB, W5, nullptr, nullptr, bufA, mTiles);
  zero_async(out, N * 3, stream);
  scatter_kernel<false><<<(E + 255) / 256, 256, 0, stream>>>(bufA, src, dst, nsrc, out, E, 3, 1);
  final_bias_kernel<<<(N * 3 + 255) / 256, 256, 0, stream>>>(out, ndst, b5, N, 3);
}